// LSTM_26388279066724
// MI455X (gfx1250) — compile-verified
//
#include <hip/hip_runtime.h>

typedef __attribute__((ext_vector_type(2))) float v2f;
typedef __attribute__((ext_vector_type(8))) float v8f;

constexpr int S_LEN = 2048;
constexpr int B_N   = 2048;
constexpr int HID   = 20;
constexpr int NT    = 8;   // 4 gates x 32 padded cols / 16-wide N tiles
constexpr int KT    = 5;   // 5 K-steps of 4 cover H=20 exactly

__device__ __forceinline__ float sigm(float v) {
    // 1 / (1 + e^-v): v_exp_f32 + v_rcp_f32
    return __frcp_rn(1.0f + __expf(-v));
}
__device__ __forceinline__ float tanh_fast(float v) {
    // tanh(x) = 2*sigmoid(2x) - 1
    return fmaf(2.0f, sigm(2.0f * v), -1.0f);
}

__global__ __launch_bounds__(32)
void lstm_wmma_kernel(const float* __restrict__ x,
                      const float* __restrict__ Wih,
                      const float* __restrict__ Whh,
                      const float* __restrict__ bih,
                      const float* __restrict__ bhh,
                      float* __restrict__ out)
{
    // h feedback buffer: 16 batch rows x 32 padded hidden cols (only 0..19 read back)
    __shared__ float h_lds[16 * 32];

    const int lane   = threadIdx.x;      // 0..31 (wave32)
    const int half   = lane >> 4;        // 0 or 1
    const int ln     = lane & 15;
    const int m_base = blockIdx.x * 16;  // 16 batch rows per wave

    // ---- loop-invariant per-lane setup --------------------------------
    // Padded gate column for N-tile t: gate group = t/2, col-in-gate = (t&1)*16 + ln
    float w0[NT], w1[NT], bb[NT];
    v2f   Bf[NT][KT];
#pragma unroll
    for (int t = 0; t < NT; ++t) {
        const int grp   = t >> 1;
        const int colp  = ((t & 1) << 4) + ln;     // 0..31 within gate
        const bool ok   = colp < HID;
        const int col   = grp * HID + (ok ? colp : 0);  // unpadded gate column
        w0[t] = ok ? Wih[col * 2 + 0] : 0.0f;
        w1[t] = ok ? Wih[col * 2 + 1] : 0.0f;
        bb[t] = ok ? (bih[col] + bhh[col]) : 0.0f;
#pragma unroll
        for (int k = 0; k < KT; ++k) {
            // B fragment (K x N = 4 x 16): lane holds N, VGPR v + lane-half hold K
            const int k0 = 4 * k + 2 * half;       // absolute K for v=0
            v2f b;
            b.x = ok ? Whh[col * HID + k0 + 0] : 0.0f;
            b.y = ok ? Whh[col * HID + k0 + 1] : 0.0f;
            Bf[t][k] = b;
        }
    }

    // cell state: [p][r] -> hidden idx ln+16p, batch row m_base + r + 8*half
    float cst[2][8];
#pragma unroll
    for (int p = 0; p < 2; ++p)
#pragma unroll
        for (int r = 0; r < 8; ++r) cst[p][r] = 0.0f;

    // ---- sequential recurrence ----------------------------------------
    for (int s = 0; s < S_LEN; ++s) {
        // x tile for this wave's 16 rows (I=2 contiguous floats per row)
        const float* xs = x + ((size_t)s * B_N + m_base + 8 * half) * 2;
        v2f xv[8];
#pragma unroll
        for (int r = 0; r < 8; ++r) xv[r] = *(const v2f*)(xs + r * 2);
        if (s + 1 < S_LEN)
            __builtin_prefetch(x + ((size_t)(s + 1) * B_N + m_base) * 2, 0, 1);

        // accumulators initialized with x @ W_ih^T + b (fused input projection)
        v8f acc[NT];
#pragma unroll
        for (int t = 0; t < NT; ++t)
#pragma unroll
            for (int r = 0; r < 8; ++r)
                acc[t][r] = fmaf(xv[r].y, w1[t], fmaf(xv[r].x, w0[t], bb[t]));

        if (s > 0) {
            // A fragments (M x K = 16 x 4) from previous h in LDS:
            // lane holds M=ln, VGPR v + lane-half hold K = 4k + v + 2*half
            v2f Af[KT];
#pragma unroll
            for (int k = 0; k < KT; ++k)
                Af[k] = *(const v2f*)&h_lds[ln * 32 + 4 * k + 2 * half];

            // gates += h @ W_hh^T : 40 x v_wmma_f32_16x16x4_f32
#pragma unroll
            for (int k = 0; k < KT; ++k)
#pragma unroll
                for (int t = 0; t < NT; ++t)
                    acc[t] = __builtin_amdgcn_wmma_f32_16x16x4_f32(
                        false, Af[k], false, Bf[t][k],
                        (short)0, acc[t], false, false);
        }

        // ---- gate nonlinearities + state update (pure per-lane) -------
#pragma unroll
        for (int p = 0; p < 2; ++p) {
#pragma unroll
            for (int r = 0; r < 8; ++r) {
                const float ig = sigm(acc[0 + p][r]);
                const float fg = sigm(acc[2 + p][r]);
                const float gg = tanh_fast(acc[4 + p][r]);
                const float og = sigm(acc[6 + p][r]);
                const float c  = fmaf(fg, cst[p][r], ig * gg);
                cst[p][r] = c;
                const float h = og * tanh_fast(c);

                // feedback for next step (cols 20..31 written but never read)
                h_lds[(r + 8 * half) * 32 + ln + 16 * p] = h;

                // emit h_t (valid hidden idx only)
                if (p == 0 || ln < 4) {
                    const size_t row = (size_t)s * B_N + m_base + r + 8 * half;
                    out[row * HID + ln + 16 * p] = h;
                }
            }
        }
        // single wave per block: DS ops are in-order within the wave, and the
        // compiler inserts s_wait_dscnt for the next iteration's A loads.
    }
}

extern "C" void kernel_launch(void* const* d_in, const int* in_sizes, int n_in,
                              void* d_out, int out_size, void* d_ws, size_t ws_size,
                              hipStream_t stream) {
    (void)in_sizes; (void)n_in; (void)out_size; (void)d_ws; (void)ws_size;
    const float* x   = (const float*)d_in[0];
    const float* Wih = (const float*)d_in[1];
    const float* Whh = (const float*)d_in[2];
    const float* bih = (const float*)d_in[3];
    const float* bhh = (const float*)d_in[4];
    float* out = (float*)d_out;

    dim3 grid(B_N / 16);   // 128 independent waves
    dim3 block(32);        // one wave32 per workgroup
    lstm_wmma_kernel<<<grid, block, 0, stream>>>(x, Wih, Whh, bih, bhh, out);
}